// LinearChainCRF_21534966022364
// MI455X (gfx1250) — compile-verified
//
#include <hip/hip_runtime.h>

// ---------------------------------------------------------------------------
// Linear-chain CRF forward (logZ) + scores tensor, for MI455X (gfx1250).
//
// scores: broadcast add, 151MB of NT b128 stores -> bandwidth bound.
// recurrence: log-semiring matvec in exp domain:
//   P(48x32) = exp(trans)(48x48) @ exp2((alpha-m)*log2e)(48x32)
// via V_WMMA_F32_16X16X4_F32 (6 tiles/step, K=48 -> 12 chained wmma per wave),
// raw v_exp_f32/v_log_f32, per-wave row-max partials, and feature staging via
// GLOBAL_LOAD_ASYNC_TO_LDS_B32 (ASYNCcnt) double-buffered across steps.
// ---------------------------------------------------------------------------

typedef __attribute__((ext_vector_type(2))) float v2f;
typedef __attribute__((ext_vector_type(4))) float v4f;
typedef __attribute__((ext_vector_type(8))) float v8f;

#define B_ 32
#define L_ 512
#define T_ 48
#define START_ID 46
#define END_ID 47

#define EPITCH 50   // E rows:   8B-aligned, 50L mod 64 distinct -> conflict-free
#define APITCH 49   // alpha/last rows
#define VPITCH 50   // V rows [b][j]: 8B-aligned so B-frag pairs -> ds_load_b64
#define FPITCH 49   // staged features

#define LOG2E 1.4426950408889634f
#define LN2   0.6931471805599453f

#if defined(__has_builtin)
#if __has_builtin(__builtin_amdgcn_global_load_async_to_lds_b32)
#define HAVE_ASYNC_LDS 1
#endif
#endif

typedef __attribute__((address_space(1))) int gint_t;
typedef __attribute__((address_space(3))) int lint_t;

// async global->LDS b32 copy (per-lane), tracked by ASYNCcnt
__device__ __forceinline__ void feat_copy_b32(const float* g, float* l) {
#ifdef HAVE_ASYNC_LDS
    __builtin_amdgcn_global_load_async_to_lds_b32(
        (gint_t*)(uintptr_t)g,
        (lint_t*)(unsigned)(uintptr_t)l,
        0, 0);
#else
    *l = *g;   // synchronous fallback
#endif
}
__device__ __forceinline__ void feat_copy_wait() {
#ifdef HAVE_ASYNC_LDS
#if __has_builtin(__builtin_amdgcn_s_wait_asynccnt)
    __builtin_amdgcn_s_wait_asynccnt(0);
#else
    asm volatile("s_wait_asynccnt 0x0" ::: "memory");
#endif
#endif
}

// ---------------------------------------------------------------------------
// Kernel A: scores[b,l,i,j] = trans[i,j] + feat[b,l,i]
// One block per (b,l): local flat p in [0,2304), trans idx == p, feat idx p/48.
// Head 3 scalars + 575 NT float4 stores + 1 tail (d_out+1 alignment fix).
// ---------------------------------------------------------------------------
__global__ void __launch_bounds__(192)
crf_scores_kernel(const float* __restrict__ feat,
                  const float* __restrict__ trans,
                  float* __restrict__ out1 /* = d_out + 1 */) {
    const int bl   = blockIdx.x;              // b*L + l
    const int base = bl * (T_ * T_);
    const float* frow = feat + bl * T_;
    float* o = out1 + base;                   // o + p, p=3+4k is 16B aligned
    const int tid = threadIdx.x;

    if (tid < 3) {
        o[tid] = trans[tid] + frow[0];                       // p=0,1,2 -> i=0
    } else if (tid == 3) {
        o[T_ * T_ - 1] = trans[T_ * T_ - 1] + frow[T_ - 1];  // p=2303 -> i=47
    }
    for (int k = tid; k < 575; k += 192) {
        const int p0 = 3 + 4 * k;
        v4f v;
        v.x = trans[p0]     + frow[p0 / T_];
        v.y = trans[p0 + 1] + frow[(p0 + 1) / T_];
        v.z = trans[p0 + 2] + frow[(p0 + 2) / T_];
        v.w = trans[p0 + 3] + frow[(p0 + 3) / T_];
        __builtin_nontemporal_store(v, reinterpret_cast<v4f*>(o + p0));
    }
}

// ---------------------------------------------------------------------------
// Kernel B: whole recurrence in one workgroup (192 threads = 6 wave32).
// Wave w owns output tile (tm = w%3 rows of i, tn = w/3 cols of b).
// ---------------------------------------------------------------------------
__global__ void __launch_bounds__(192)
crf_forward_kernel(const float* __restrict__ feat,
                   const float* __restrict__ trans,
                   const int* __restrict__ seqLens,
                   float* __restrict__ logZ_out) {
    __shared__ __align__(16) float Eexp[T_ * EPITCH];      // exp(trans[i][j])
    __shared__ __align__(16) float alphaS[B_ * APITCH];    // alpha[b][i]
    __shared__ __align__(16) float Vs[B_ * VPITCH];        // V[b][j] (B-frag source)
    __shared__ __align__(16) float featS[2][B_ * FPITCH];  // double-buffered features
    __shared__ float pmS[3 * B_];                          // per-(tm,b) row-max partials
    __shared__ float mS[B_];                               // finalized per-batch max
    __shared__ __align__(16) float lastS[B_ * APITCH];     // alpha at t = seqLen-1
    __shared__ int   lenS[B_];
    __shared__ float lseS[B_];

    const int tid  = threadIdx.x;
    const int lane = tid & 31;
    const int wave = tid >> 5;
    const int tm   = wave % 3;          // i-tile: rows tm*16 .. tm*16+15
    const int tn   = wave / 3;          // b-tile: cols tn*16 .. tn*16+15
    const int half = lane >> 4;         // K-half select (ISA f32 A/B layout)
    const int l16  = lane & 15;

    // ---- init: E = exp(trans), seqLens, alpha0, stage feat[t=1] ------------
    for (int e = tid; e < T_ * T_; e += 192) {
        int i = e / T_, j = e % T_;
        Eexp[i * EPITCH + j] = __builtin_amdgcn_exp2f(trans[e] * LOG2E);
    }
    if (tid < B_) lenS[tid] = seqLens[tid];
    #pragma unroll
    for (int k = 0; k < 8; ++k) {                    // 1536 = 8*192, exact
        int e = tid + 192 * k;
        int b = e / T_, i = e % T_;
        alphaS[b * APITCH + i] = trans[i * T_ + START_ID] + feat[b * (L_ * T_) + i];
        featS[1][b * FPITCH + i] = feat[(b * L_ + 1) * T_ + i];
    }
    __syncthreads();
    // pm init (partial 0 = serial max of alpha0; 1,2 = -inf); seqLen==1 capture
    if (tid < B_) {
        float mx = -3.4e38f;
        for (int j = 0; j < T_; ++j) mx = fmaxf(mx, alphaS[tid * APITCH + j]);
        pmS[tid] = mx;
    } else if (tid < 96) {
        pmS[tid] = -3.4e38f;
    }
    #pragma unroll
    for (int k = 0; k < 8; ++k) {
        int e = tid + 192 * k;
        int b = e / T_, i = e % T_;
        if (lenS[b] == 1) lastS[b * APITCH + i] = alphaS[b * APITCH + i];
    }
    __syncthreads();

    // loop-invariant per-thread values
    const int mrow = tm * 16 + l16;        // A-frag: lane -> matrix row M
    const int ncol = tn * 16 + l16;        // B-frag: lane -> matrix col N
    const int bcol = ncol;                 // C/D frag: lane -> col (batch)
    const int tcap = lenS[bcol] - 1;       // capture step for this batch
    const int bv   = tid & 31;             // Phase-A batch
    const int jv0  = tid >> 5;             // Phase-A first j

    // ---- time recurrence: 2 barriers / step ------------------------------
    #pragma unroll 2
    for (int t = 1; t < L_; ++t) {
        const int p = t & 1;

        // Phase A: finalize m[b] from 3 partials; V[b][j] = exp(alpha - m)
        {
            const float mb = fmaxf(fmaxf(pmS[bv], pmS[32 + bv]), pmS[64 + bv]);
            if (tid < 32) mS[bv] = mb;
            #pragma unroll
            for (int k = 0; k < 6; ++k) {
                const int j = jv0 + 6 * k;
                Vs[bv * VPITCH + j] =
                    __builtin_amdgcn_exp2f((alphaS[bv * APITCH + j] - mb) * LOG2E);
            }
        }
        __syncthreads();

        // Phase B: async-prefetch features for t+1 (overlaps with wmma chain)
        if (t + 1 < L_) {
            #pragma unroll
            for (int k = 0; k < 8; ++k) {
                int e = tid + 192 * k;
                int b = e / T_, i = e % T_;
                feat_copy_b32(&feat[(b * L_ + t + 1) * T_ + i],
                              &featS[p ^ 1][b * FPITCH + i]);
            }
        }

        // P(16x16 tile) = E_tile @ V_tile via 12 chained f32 WMMAs (K=48)
        v8f c = {};
        #pragma unroll
        for (int k0 = 0; k0 < T_; k0 += 4) {
            const int ka = k0 + half * 2;   // lanes 16-31 carry K+2 (ISA layout)
            v2f a, bf;
            a.x  = Eexp[mrow * EPITCH + ka];
            a.y  = Eexp[mrow * EPITCH + ka + 1];
            bf.x = Vs[ncol * VPITCH + ka];      // consecutive -> ds_load_b64
            bf.y = Vs[ncol * VPITCH + ka + 1];
            c = __builtin_amdgcn_wmma_f32_16x16x4_f32(
                    false, a, false, bf, (short)0, c, false, false);
        }

        // alpha_new[b][i] = feat + m_b + ln2*log2(P); row-max; single capture
        const float mb = mS[bcol];
        float avr[8];
        float rmax = -3.4e38f;
        #pragma unroll
        for (int r = 0; r < 8; ++r) {
            const int i = tm * 16 + half * 8 + r;   // C/D frag: vgpr r -> row
            avr[r] = featS[p][bcol * FPITCH + i] + mb +
                     __builtin_amdgcn_logf(c[r]) * LN2;
            alphaS[bcol * APITCH + i] = avr[r];
            rmax = fmaxf(rmax, avr[r]);
        }
        if (t == tcap) {                            // one exec toggle, 8 stores
            #pragma unroll
            for (int r = 0; r < 8; ++r)
                lastS[bcol * APITCH + tm * 16 + half * 8 + r] = avr[r];
        }
        rmax = fmaxf(rmax, __shfl_xor(rmax, 16, 32));   // combine halves
        if (half == 0) pmS[tm * 32 + bcol] = rmax;

        feat_copy_wait();          // retire async copies before the barrier
        __syncthreads();
    }

    // ---- finalize: logZ = sum_b LSE_i( last[b][i] + trans[END][i] ) --------
    if (tid < B_) {
        float mx = -3.4e38f;
        for (int i = 0; i < T_; ++i)
            mx = fmaxf(mx, lastS[tid * APITCH + i] + trans[END_ID * T_ + i]);
        float s = 0.f;
        for (int i = 0; i < T_; ++i)
            s += __builtin_amdgcn_exp2f(
                     (lastS[tid * APITCH + i] + trans[END_ID * T_ + i] - mx) * LOG2E);
        lseS[tid] = mx + __builtin_amdgcn_logf(s) * LN2;
    }
    __syncthreads();
    if (tid == 0) {
        float z = 0.f;
        for (int b = 0; b < B_; ++b) z += lseS[b];
        logZ_out[0] = z;
    }
}

// ---------------------------------------------------------------------------
extern "C" void kernel_launch(void* const* d_in, const int* in_sizes, int n_in,
                              void* d_out, int out_size, void* d_ws, size_t ws_size,
                              hipStream_t stream) {
    const float* feat    = (const float*)d_in[0];   // (B,L,T) f32
    const float* trans   = (const float*)d_in[1];   // (T,T)   f32
    const int*   seqLens = (const int*)d_in[2];     // (B,)    i32
    // d_in[3] = mask, derived from seqLens, unused.

    float* out = (float*)d_out;          // out[0] = logZ, out[1..] = scores flat

    crf_scores_kernel<<<B_ * L_, 192, 0, stream>>>(feat, trans, out + 1);
    crf_forward_kernel<<<1, 192, 0, stream>>>(feat, trans, seqLens, out);
}